// VectorQuantizer_1125281431598
// MI455X (gfx1250) — compile-verified
//
#include <hip/hip_runtime.h>
#include <hip/hip_bf16.h>

typedef __attribute__((ext_vector_type(2))) float v2f;
typedef __attribute__((ext_vector_type(8))) float v8f;
typedef __attribute__((ext_vector_type(4))) int v4i;

#define B_SZ 32768
#define K_SZ 8192
#define D_SZ 512
#define LDS_STRIDE 36  // pad 32 -> 36 so 16 rows hit distinct bank groups

#if __has_builtin(__builtin_amdgcn_global_load_async_to_lds_b128)
#define HAS_ASYNC_LDS 1
#else
#define HAS_ASYNC_LDS 0
#endif

__device__ __forceinline__ void wait_async_copies() {
#if __has_builtin(__builtin_amdgcn_s_wait_asynccnt)
  __builtin_amdgcn_s_wait_asynccnt(0);
#else
  asm volatile("s_wait_asynccnt 0x0" ::: "memory");
#endif
}

#if HAS_ASYNC_LDS
typedef __attribute__((address_space(1))) v4i gas_v4i;  // global int4
typedef __attribute__((address_space(3))) v4i las_v4i;  // LDS int4

__device__ __forceinline__ void async_cp16(const float* g, float* l) {
  __builtin_amdgcn_global_load_async_to_lds_b128((gas_v4i*)g, (las_v4i*)l, 0, 0);
}
#endif

// Order-preserving float->uint mapping (monotonic for all finite floats)
__device__ __forceinline__ unsigned int ford(float f) {
  unsigned int u = __float_as_uint(f);
  return (u & 0x80000000u) ? ~u : (u | 0x80000000u);
}

__global__ void vq_init(unsigned long long* __restrict__ best) {
  int i = blockIdx.x * blockDim.x + threadIdx.x;
  if (i < B_SZ) best[i] = ~0ull;
}

// ||e_k||^2 for each codebook row: one block (128 threads) per row
__global__ void vq_wnorm(const float* __restrict__ W, float* __restrict__ wnorm) {
  const int k = blockIdx.x;
  const int t = threadIdx.x;
  float4 v = *(const float4*)&W[(size_t)k * D_SZ + t * 4];
  float s = v.x * v.x + v.y * v.y + v.z * v.z + v.w * v.w;
  for (int m = 16; m; m >>= 1) s += __shfl_down(s, m, 32);
  __shared__ float red[4];
  if ((t & 31) == 0) red[t >> 5] = s;
  __syncthreads();
  if (t == 0) wnorm[k] = red[0] + red[1] + red[2] + red[3];
}

// Tiled fp32 WMMA GEMM (S = X * W^T) fused with per-row argmin of
// dist = ||e_k||^2 - 2*S[b,k], merged globally via u64 atomicMin.
// Double-buffered LDS tiles filled via async global->LDS copies, and
// double-buffered WMMA fragments to hide DS latency.
__global__ __launch_bounds__(256) void vq_dist_argmin(
    const float* __restrict__ X, const float* __restrict__ W,
    const float* __restrict__ wnorm, unsigned long long* __restrict__ best) {
  __shared__ float Xs[2][128 * LDS_STRIDE];
  __shared__ float Ws[2][128 * LDS_STRIDE];

  const int m0 = blockIdx.y * 128;  // batch rows
  const int k0 = blockIdx.x * 128;  // codebook rows
  const int tid = threadIdx.x;
  const int lane = tid & 31;
  const int wave = tid >> 5;
  const int wm = (wave >> 2) * 64;  // 2 waves in M, each 64 rows
  const int wn = (wave & 3) * 32;   // 4 waves in N, each 32 cols
  const int l15 = lane & 15;
  const int hb = lane >> 4;         // half-wave select
  const int aoff = hb * 2;          // inner-K offset per ISA A/B layout

  const int lr = tid >> 3;        // 0..31: row within tile slice
  const int lc = (tid & 7) << 2;  // 0,4,...,28: float4 column

  // Issue the global->LDS copy of the 128x32 X and W tiles at depth d0.
  auto issue_copy = [&](int buf, int d0) {
#pragma unroll
    for (int r = 0; r < 128; r += 32) {
#if HAS_ASYNC_LDS
      async_cp16(&X[(size_t)(m0 + lr + r) * D_SZ + d0 + lc],
                 &Xs[buf][(lr + r) * LDS_STRIDE + lc]);
      async_cp16(&W[(size_t)(k0 + lr + r) * D_SZ + d0 + lc],
                 &Ws[buf][(lr + r) * LDS_STRIDE + lc]);
#else
      *(float4*)&Xs[buf][(lr + r) * LDS_STRIDE + lc] =
          *(const float4*)&X[(size_t)(m0 + lr + r) * D_SZ + d0 + lc];
      *(float4*)&Ws[buf][(lr + r) * LDS_STRIDE + lc] =
          *(const float4*)&W[(size_t)(k0 + lr + r) * D_SZ + d0 + lc];
#endif
    }
  };

  v8f acc[4][2];
  const v8f vzero = {0.f, 0.f, 0.f, 0.f, 0.f, 0.f, 0.f, 0.f};
#pragma unroll
  for (int s = 0; s < 4; ++s)
#pragma unroll
    for (int t = 0; t < 2; ++t) acc[s][t] = vzero;

  issue_copy(0, 0);
#if HAS_ASYNC_LDS
  wait_async_copies();
#endif
  __syncthreads();

  for (int it = 0; it < 16; ++it) {
    const int cur = it & 1;
    if (it < 15) issue_copy(cur ^ 1, (it + 1) * 32);  // overlap with compute

    const float* xb = &Xs[cur][0];
    const float* wb = &Ws[cur][0];

    // Fragment double buffer: prefetch step+1 while WMMAs of step execute.
    v2f a[2][4], bf[2][2];
#pragma unroll
    for (int s = 0; s < 4; ++s)
      a[0][s] = *(const v2f*)&xb[(wm + s * 16 + l15) * LDS_STRIDE + aoff];
#pragma unroll
    for (int t = 0; t < 2; ++t)
      bf[0][t] = *(const v2f*)&wb[(wn + t * 16 + l15) * LDS_STRIDE + aoff];

#pragma unroll
    for (int step = 0; step < 8; ++step) {
      const int fc = step & 1, fn = fc ^ 1;
      if (step < 7) {
        const int dd = (step + 1) * 4;
#pragma unroll
        for (int s = 0; s < 4; ++s)
          a[fn][s] =
              *(const v2f*)&xb[(wm + s * 16 + l15) * LDS_STRIDE + dd + aoff];
#pragma unroll
        for (int t = 0; t < 2; ++t)
          bf[fn][t] =
              *(const v2f*)&wb[(wn + t * 16 + l15) * LDS_STRIDE + dd + aoff];
      }
#pragma unroll
      for (int s = 0; s < 4; ++s)
#pragma unroll
        for (int t = 0; t < 2; ++t)
          acc[s][t] = __builtin_amdgcn_wmma_f32_16x16x4_f32(
              false, a[fc][s], false, bf[fc][t], (short)0, acc[s][t], false,
              false);
    }

#if HAS_ASYNC_LDS
    if (it < 15) wait_async_copies();
#endif
    __syncthreads();
  }

  // Epilogue: dist = ||e||^2 - 2*S ; pack (ordered_dist, k) and argmin-reduce.
  float nrm[2];
#pragma unroll
  for (int t = 0; t < 2; ++t) nrm[t] = wnorm[k0 + wn + t * 16 + l15];

#pragma unroll
  for (int s = 0; s < 4; ++s) {
#pragma unroll
    for (int i = 0; i < 8; ++i) {
      unsigned long long p = ~0ull;
#pragma unroll
      for (int t = 0; t < 2; ++t) {
        float dval = nrm[t] - 2.0f * acc[s][t][i];
        unsigned long long c = ((unsigned long long)ford(dval) << 32) |
                               (unsigned int)(k0 + wn + t * 16 + l15);
        p = c < p ? c : p;
      }
      // min across the 16 lanes of each half-wave (halves are independent rows)
#pragma unroll
      for (int m = 1; m <= 8; m <<= 1) {
        unsigned long long o = __shfl_xor(p, m, 32);
        p = o < p ? o : p;
      }
      if (l15 == 0) {
        int row = m0 + wm + s * 16 + i + hb * 8;  // C layout: lanes>=16 are M+8
        atomicMin(&best[row], p);
      }
    }
  }
}

// Gather codebook rows, write quantized output + indices, per-row SSE partial.
__global__ void vq_gather(const float* __restrict__ X, const float* __restrict__ W,
                          const unsigned long long* __restrict__ best,
                          float* __restrict__ outq, int* __restrict__ idx_out,
                          float* __restrict__ partial) {
  const int b = blockIdx.x;
  const int t = threadIdx.x;  // 128
  const unsigned int idx = (unsigned int)(best[b] & 0xFFFFFFFFull);
  float4 q = *(const float4*)&W[(size_t)idx * D_SZ + t * 4];
  float4 x = *(const float4*)&X[(size_t)b * D_SZ + t * 4];
  *(float4*)&outq[(size_t)b * D_SZ + t * 4] = q;
  float dx = q.x - x.x, dy = q.y - x.y, dz = q.z - x.z, dw = q.w - x.w;
  float s = dx * dx + dy * dy + dz * dz + dw * dw;
  for (int m = 16; m; m >>= 1) s += __shfl_down(s, m, 32);
  __shared__ float red[4];
  if ((t & 31) == 0) red[t >> 5] = s;
  __syncthreads();
  if (t == 0) {
    partial[b] = red[0] + red[1] + red[2] + red[3];
    idx_out[b] = (int)idx;
  }
}

// Deterministic fixed-order final loss reduction.
__global__ void vq_finalize(const float* __restrict__ partial,
                            float* __restrict__ loss_out) {
  const int t = threadIdx.x;  // 256
  float s = 0.f;
  for (int i = t; i < B_SZ; i += 256) s += partial[i];
  for (int m = 16; m; m >>= 1) s += __shfl_down(s, m, 32);
  __shared__ float red[8];
  if ((t & 31) == 0) red[t >> 5] = s;
  __syncthreads();
  if (t == 0) {
    float tot = 0.f;
#pragma unroll
    for (int i = 0; i < 8; ++i) tot += red[i];
    *loss_out = 1.9f * tot / ((float)B_SZ * (float)D_SZ);
  }
}

extern "C" void kernel_launch(void* const* d_in, const int* in_sizes, int n_in,
                              void* d_out, int out_size, void* d_ws, size_t ws_size,
                              hipStream_t stream) {
  const float* X = (const float*)d_in[0];  // [B, D]
  const float* W = (const float*)d_in[1];  // [K, D]
  float* out = (float*)d_out;

  char* ws = (char*)d_ws;
  unsigned long long* best = (unsigned long long*)ws;                     // B * 8
  float* wnorm = (float*)(ws + (size_t)B_SZ * 8);                         // K * 4
  float* partial = (float*)(ws + (size_t)B_SZ * 8 + (size_t)K_SZ * 4);    // B * 4

  float* outq = out;                                    // [B*D] quantized_st
  float* loss_out = out + (size_t)B_SZ * D_SZ;          // [1]  loss
  int* idx_out = (int*)(out + (size_t)B_SZ * D_SZ + 1); // [B]  indices (int32)

  hipLaunchKernelGGL(vq_init, dim3(B_SZ / 256), dim3(256), 0, stream, best);
  hipLaunchKernelGGL(vq_wnorm, dim3(K_SZ), dim3(128), 0, stream, W, wnorm);
  hipLaunchKernelGGL(vq_dist_argmin, dim3(K_SZ / 128, B_SZ / 128), dim3(256), 0,
                     stream, X, W, wnorm, best);
  hipLaunchKernelGGL(vq_gather, dim3(B_SZ), dim3(128), 0, stream, X, W, best,
                     outq, idx_out, partial);
  hipLaunchKernelGGL(vq_finalize, dim3(1), dim3(256), 0, stream, partial, loss_out);
}